// Model_47820165873826
// MI455X (gfx1250) — compile-verified
//
#include <hip/hip_runtime.h>

typedef __attribute__((ext_vector_type(2))) float v2f;
typedef __attribute__((ext_vector_type(8))) float v8f;

// Hardware tanh (CDNA5 TRANS op). ISA §7.4: one independent op needed after a
// TRANS op before its result may be consumed -> pad with a single v_nop.
__device__ __forceinline__ float tanh_hw(float x) {
    float y;
    asm volatile("v_tanh_f32 %0, %1\n\tv_nop" : "=v"(y) : "v"(x));
    return y;
}

// ---------------------------------------------------------------------------
// Kernel 1: xp[row] = dot(x[row, 0:128], W) + bias  via V_WMMA_F32_16X16X4_F32
// Each wave computes 16 rows. B-matrix columns are all broadcast copies of W,
// so every column of D holds the desired dot products. Memory-bound: streams
// 256 MB of x once (~11 us at 23.3 TB/s).
// ---------------------------------------------------------------------------
__global__ void __launch_bounds__(256) rnn_proj_wmma(
    const float* __restrict__ x, const float* __restrict__ W,
    const float* __restrict__ b_ih0, const float* __restrict__ b_hh0,
    float* __restrict__ xp, long rows)
{
    const int lane = threadIdx.x & 31;
    const int wave = threadIdx.x >> 5;
    const long rowBase = ((long)blockIdx.x * 8 + wave) * 16;
    if (rowBase >= rows) return;                 // wave-uniform: EXEC stays all-1s

    const int half = lane >> 4;                  // 0: K=0,1 slots  1: K=2,3 slots
    const int m    = lane & 15;                  // A-matrix row for this lane
    const float bias = b_ih0[0] + b_hh0[0];
    const float* xrow = x + (rowBase + m) * 128;

    v8f c;
    #pragma unroll
    for (int j = 0; j < 8; ++j) c[j] = bias;     // fold bias into the accumulator

    #pragma unroll
    for (int i = 0; i < 32; ++i) {               // 32 x (K=4) = K=128
        const int k0 = 4 * i + 2 * half;
        v2f a, b;
        a.x = xrow[k0];  a.y = xrow[k0 + 1];     // A slot (m, k0..k0+1)
        b.x = W[k0];     b.y = W[k0 + 1];        // B slot with identical K map
        c = __builtin_amdgcn_wmma_f32_16x16x4_f32(
                /*neg_a=*/false, a, /*neg_b=*/false, b,
                /*c_mod=*/(short)0, c, /*reuse_a=*/false, /*reuse_b=*/false);
    }

    // D layout: lane L<16 VGPR v = D[M=v][N=L]; lane>=16 VGPR v = D[M=v+8][N=L-16].
    // All N columns identical -> lane 0 holds rows 0..7, lane 16 rows 8..15.
    if (m == 0) {
        float4* p = (float4*)(xp + rowBase + half * 8);
        p[0] = make_float4(c[0], c[1], c[2], c[3]);
        p[1] = make_float4(c[4], c[5], c[6], c[7]);
    }
}

// ---------------------------------------------------------------------------
// Kernel 2: serial 2-layer H=1 recurrence, latency-bound on the tanh chain.
// Re-indexed so each iteration computes h1_s and h2_{s-1}: both depend only on
// previous-iteration state, so their v_tanh_f32 latencies overlap and the
// critical path is ~one fma+tanh per step (vs two when computed in-order).
// One lane per batch chain; 8 blocks of 32 threads so each wave gets its own
// WGP (private TRANS unit).
// ---------------------------------------------------------------------------
__global__ void __launch_bounds__(32) rnn_scan(
    const float* __restrict__ xp,
    const float* __restrict__ W_hh0, const float* __restrict__ W_ih1,
    const float* __restrict__ W_hh1, const float* __restrict__ b_ih1,
    const float* __restrict__ b_hh1, float* __restrict__ out,
    int S, int B)
{
    const int b = blockIdx.x * 32 + (int)threadIdx.x;
    if (b >= B) return;
    const float w0 = W_hh0[0];
    const float wi = W_ih1[0];
    const float wh = W_hh1[0];
    const float b1 = b_ih1[0] + b_hh1[0];
    const float* p = xp + b;

    float h1 = tanh_hw(p[0]);                    // h1_0 (h1_{-1} = 0)
    float h2 = 0.0f;                             // h2_{-1}
    #pragma unroll 4
    for (int s = 1; s < S; ++s) {
        __builtin_prefetch(p + (long)(s + 64) * B, 0, 1);
        float v   = p[(long)s * B];
        float nh1 = tanh_hw(fmaf(h1, w0, v));                   // h1_s
        h2        = tanh_hw(fmaf(h2, wh, fmaf(h1, wi, b1)));    // h2_{s-1}
        h1        = nh1;
    }
    h2 = tanh_hw(fmaf(h2, wh, fmaf(h1, wi, b1)));               // h2_{S-1}
    out[b] = 1.0f / (1.0f + __expf(-h2));
}

// ---------------------------------------------------------------------------
// Fallback if workspace is too small for xp: fused scalar version (slow but
// correct/deterministic).
// ---------------------------------------------------------------------------
__global__ void __launch_bounds__(32) rnn_fused_fallback(
    const float* __restrict__ x, const float* __restrict__ W,
    const float* __restrict__ W_hh0, const float* __restrict__ b_ih0,
    const float* __restrict__ b_hh0, const float* __restrict__ W_ih1,
    const float* __restrict__ W_hh1, const float* __restrict__ b_ih1,
    const float* __restrict__ b_hh1, float* __restrict__ out,
    int S, int B, int I)
{
    const int b = blockIdx.x * 32 + (int)threadIdx.x;
    if (b >= B) return;
    const float bias0 = b_ih0[0] + b_hh0[0];
    const float w0 = W_hh0[0];
    const float wi = W_ih1[0];
    const float wh = W_hh1[0];
    const float b1 = b_ih1[0] + b_hh1[0];
    float h1 = 0.0f, h2 = 0.0f;
    for (int s = 0; s < S; ++s) {
        const float* xr = x + ((long)s * B + b) * I;
        float acc = bias0;
        for (int k = 0; k < I; ++k) acc = fmaf(xr[k], W[k], acc);
        h1 = tanh_hw(fmaf(h1, w0, acc));
        h2 = tanh_hw(fmaf(h2, wh, fmaf(h1, wi, b1)));
    }
    out[b] = 1.0f / (1.0f + __expf(-h2));
}

extern "C" void kernel_launch(void* const* d_in, const int* in_sizes, int n_in,
                              void* d_out, int out_size, void* d_ws, size_t ws_size,
                              hipStream_t stream) {
    const float* x     = (const float*)d_in[0];
    const float* W_ih0 = (const float*)d_in[1];
    const float* W_hh0 = (const float*)d_in[2];
    const float* b_ih0 = (const float*)d_in[3];
    const float* b_hh0 = (const float*)d_in[4];
    const float* W_ih1 = (const float*)d_in[5];
    const float* W_hh1 = (const float*)d_in[6];
    const float* b_ih1 = (const float*)d_in[7];
    const float* b_hh1 = (const float*)d_in[8];
    float* out = (float*)d_out;

    const int I = in_sizes[1];                  // W_ih0 is [1, I] -> 128
    const int B = out_size;                     // 256
    const int S = in_sizes[0] / (B * I);        // 2048
    const long rows = (long)S * B;              // 524288

    const size_t need = (size_t)rows * sizeof(float);
    if (ws_size >= need) {
        float* xp = (float*)d_ws;
        const int rowsPerBlock = 8 * 16;        // 8 waves x 16 rows
        const int grid = (int)((rows + rowsPerBlock - 1) / rowsPerBlock);
        rnn_proj_wmma<<<grid, 256, 0, stream>>>(x, W_ih0, b_ih0, b_hh0, xp, rows);
        rnn_scan<<<(B + 31) / 32, 32, 0, stream>>>(xp, W_hh0, W_ih1, W_hh1,
                                                   b_ih1, b_hh1, out, S, B);
    } else {
        rnn_fused_fallback<<<(B + 31) / 32, 32, 0, stream>>>(
            x, W_ih0, W_hh0, b_ih0, b_hh0, W_ih1, W_hh1, b_ih1, b_hh1,
            out, S, B, I);
    }
}